// GCNN_74577812128024
// MI455X (gfx1250) — compile-verified
//
#include <hip/hip_runtime.h>

// Problem constants (from reference): B=8, N=10000, E=320000, C_IN=C_OUT=128.
#define BB 8
#define NN 10000
#define EE 320000
#define CC 128
#define LDS_STRIDE 132   // 128 + 4 pad floats -> consecutive rows shift 4 banks

typedef __attribute__((ext_vector_type(2))) float v2f;
typedef __attribute__((ext_vector_type(8))) float v8f;

// ---------------------------------------------------------------------------
// Phase 0: zero the accumulator (we accumulate ax into d_out, then transform
// in place). d_out is poisoned by the harness, so this runs every call.
// ---------------------------------------------------------------------------
__global__ __launch_bounds__(256) void GCNN_zero_kernel(float* __restrict__ p,
                                                        unsigned n4) {
    unsigned i = blockIdx.x * 256u + threadIdx.x;
    if (i < n4) {
        ((float4*)p)[i] = make_float4(0.f, 0.f, 0.f, 0.f);
    }
}

// ---------------------------------------------------------------------------
// Phase 1: COO scatter-add. One wave32 per edge; lane l handles channels
// [4l, 4l+4): float4 gather of the source row (fully coalesced, 512B/wave,
// L2-resident) and 4 global f32 atomics to the destination row.
// ---------------------------------------------------------------------------
__global__ __launch_bounds__(256) void GCNN_spmm_kernel(
    const float* __restrict__ x,     // [B, N, 128]
    const int*   __restrict__ rows,  // [B, E]
    const int*   __restrict__ cols,  // [B, E]
    const float* __restrict__ vals,  // [B, E]
    float*       __restrict__ ax)    // [B, N, 128] accumulator
{
    unsigned tid  = blockIdx.x * 256u + threadIdx.x;   // < B*E*32 = 81.92M
    unsigned e    = tid >> 5;                          // global edge idx
    unsigned lane = tid & 31u;
    if (e >= (unsigned)(BB * EE)) return;

    unsigned b   = e / (unsigned)EE;                   // batch index
    int      dst = rows[e];
    int      src = cols[e];
    float    v   = vals[e];

    const float4* xs = (const float4*)(x + ((size_t)b * NN + src) * CC);
    float4 xv = xs[lane];

    float* axp = ax + ((size_t)b * NN + dst) * CC + lane * 4u;
    atomicAdd(axp + 0, v * xv.x);
    atomicAdd(axp + 1, v * xv.y);
    atomicAdd(axp + 2, v * xv.z);
    atomicAdd(axp + 3, v * xv.w);
}

// ---------------------------------------------------------------------------
// Phase 2: out = relu(ax @ W + bias), in place on the accumulator.
// Block = 256 threads = 8 waves, owns 16 rows of the flattened [B*N, 128]
// matrix. A-tile (16x128 f32) staged in LDS (padded stride), then wave w
// computes column tile [16w, 16w+16) with V_WMMA_F32_16X16X4_F32, K stepped
// by 4 (32 chained WMMAs, f32 end-to-end => same math as reference).
// ---------------------------------------------------------------------------
__global__ __launch_bounds__(256) void GCNN_gemm_relu_kernel(
    float*       __restrict__ inout, // [B*N, 128], holds ax on entry
    const float* __restrict__ W,     // [128, 128] row-major (k, n)
    const float* __restrict__ bias)  // [128]
{
    __shared__ float Atile[16 * LDS_STRIDE];

    unsigned t    = threadIdx.x;
    size_t   base = (size_t)blockIdx.x * 16u * CC;     // first element of tile

    // Stage 16x128 A-tile: each thread moves 8 contiguous floats (2x float4).
    {
        unsigned idx = t * 8u;          // 0..2047
        unsigned r   = idx >> 7;        // row 0..15
        unsigned c   = idx & 127u;      // col, multiple of 8
        const float* gp = inout + base + (size_t)r * CC + c;
        float4 v0 = *(const float4*)(gp);
        float4 v1 = *(const float4*)(gp + 4);
        float* lp = &Atile[r * LDS_STRIDE + c];
        *(float4*)(lp)     = v0;        // (r*132 + c) % 4 == 0 -> 16B aligned
        *(float4*)(lp + 4) = v1;
    }
    __syncthreads();

    unsigned wave  = t >> 5;            // 0..7 -> column tile
    unsigned lane  = t & 31u;
    unsigned half  = lane >> 4;         // 0: lanes 0-15, 1: lanes 16-31
    unsigned l16   = lane & 15u;
    unsigned ncol0 = wave * 16u;

    v8f acc = {};
    #pragma unroll 4
    for (unsigned k0 = 0; k0 < CC; k0 += 4) {
        unsigned kk = k0 + 2u * half;   // ISA layout: VGPR0/1 = K, K+1 per half
        v2f a, bf;
        // A fragment: row M = l16, K = kk, kk+1 (contiguous -> ds b64 read)
        a.x = Atile[l16 * LDS_STRIDE + kk];
        a.y = Atile[l16 * LDS_STRIDE + kk + 1];
        // B fragment: rows K = kk, kk+1 of W, column N = ncol0 + l16
        bf.x = W[(size_t)kk * CC + ncol0 + l16];
        bf.y = W[(size_t)(kk + 1) * CC + ncol0 + l16];
        acc = __builtin_amdgcn_wmma_f32_16x16x4_f32(
            /*neg_a=*/false, a, /*neg_b=*/false, bf,
            /*c_mod=*/(short)0, acc, /*reuse_a=*/false, /*reuse_b=*/false);
    }

    // Epilogue: bias + relu, write back in place.
    float bb = bias[ncol0 + l16];
    float* outp = inout + base;
    #pragma unroll
    for (int r = 0; r < 8; ++r) {
        // D VGPR r: M = r (lanes 0-15), M = r+8 (lanes 16-31); N = l16
        float vv = acc[r] + bb;
        vv = vv > 0.f ? vv : 0.f;
        outp[(size_t)(r + 8u * half) * CC + ncol0 + l16] = vv;
    }
}

// ---------------------------------------------------------------------------
// Inputs (setup_inputs order): x, rows, cols, vals, W, b
// ---------------------------------------------------------------------------
extern "C" void kernel_launch(void* const* d_in, const int* in_sizes, int n_in,
                              void* d_out, int out_size, void* d_ws, size_t ws_size,
                              hipStream_t stream) {
    const float* x    = (const float*)d_in[0];
    const int*   rows = (const int*)  d_in[1];
    const int*   cols = (const int*)  d_in[2];
    const float* vals = (const float*)d_in[3];
    const float* W    = (const float*)d_in[4];
    const float* bias = (const float*)d_in[5];
    float*       out  = (float*)d_out;

    // Phase 0: zero accumulator (B*N*128 floats = 2,560,000 float4 stores)
    unsigned n4 = (unsigned)(BB * NN * CC) / 4u;
    GCNN_zero_kernel<<<(n4 + 255u) / 256u, 256, 0, stream>>>(out, n4);

    // Phase 1: edge scatter — B*E edges, 32 lanes each -> 320,000 blocks
    unsigned nthreads = (unsigned)(BB * EE) * 32u;
    GCNN_spmm_kernel<<<nthreads / 256u, 256, 0, stream>>>(x, rows, cols, vals, out);

    // Phase 2: in-place dense transform — 80,000 rows / 16 = 5,000 blocks
    GCNN_gemm_relu_kernel<<<(BB * NN) / 16, 256, 0, stream>>>(out, W, bias);
}